// SSIMMSE_19688130085815
// MI455X (gfx1250) — compile-verified
//
#include <hip/hip_runtime.h>

typedef float v2f __attribute__((ext_vector_type(2)));
typedef float v8f __attribute__((ext_vector_type(8)));

#define IMG      256
#define TILE_H   16
#define TILE_W   128          // 8 waves x 16 cols
#define HALO     3
#define IN_H     22           // TILE_H + 2*HALO
#define IN_STR   136          // padded LDS stride (covers 134 used cols + 2 pad)
#define NBLK     6144         // 192 planes * 16 row-strips * 2 col-strips
#define NTOT     12582912.0f  // 8*8*3*256*256

// Banded 7-tap normalized gaussian (sigma=1.5): value g[d] for d in 0..6,
// zero outside. For our d range [-15,23], in-band <=> |d-3| <= 3, so no
// explicit range check is needed (|d-3|>=4 falls through to 0).
__device__ __forceinline__ float gw0(int d) {
    int a = d - 3; a = a < 0 ? -a : a;   // |d-3|
    float r = 0.f;
    r = (a == 3) ? 0.0366329f : r;
    r = (a == 2) ? 0.1112808f : r;
    r = (a == 1) ? 0.2167454f : r;
    r = (a == 0) ? 0.2706822f : r;
    return r;
}

// wave-relative LDS byte offset of a shared-memory address: flat shared
// address is {aperture_hi, lds_offset_lo}; HW consumes addr[31:0].
__device__ __forceinline__ unsigned lds_off(const void* p) {
    return (unsigned)(uintptr_t)p;
}

__global__ __launch_bounds__(256) void ssim_tile_kernel(
    const float* __restrict__ x, const float* __restrict__ y,
    float* __restrict__ partial)
{
    __shared__ __align__(16) float xs[IN_H * IN_STR];
    __shared__ __align__(16) float ys[IN_H * IN_STR];
    __shared__ __align__(16) float hbuf[5 * 8 * 24 * 16]; // [q][wave][24][16]
    __shared__ __align__(16) float red[256];

    const int tid  = threadIdx.x;
    const int lane = tid & 31;
    const int wv   = tid >> 5;

    const int b   = blockIdx.x;
    const int p   = b >> 5;          // plane 0..191
    const int rem = b & 31;
    const int gy0 = (rem >> 1) * TILE_H;
    const int gx0 = (rem & 1) * TILE_W;

    const float* xp = x + (size_t)p * (IMG * IMG);
    const float* yp = y + (size_t)p * (IMG * IMG);

    // ---- async halo tile load: in-range lanes DMA memory->LDS directly
    // (GLOBAL_LOAD_ASYNC_TO_LDS_B32, ASYNCcnt, EXEC-masked); out-of-range
    // lanes zero-fill. Address sets are disjoint -> no DS/async hazard.
    for (int i = tid; i < IN_H * IN_STR; i += 256) {
        int r = i / IN_STR, c = i - r * IN_STR;
        int gy = gy0 - HALO + r;
        int gx = gx0 - HALO + c;
        if (gy >= 0 && gy < IMG && gx >= 0 && gx < IMG) {
            unsigned goff = (unsigned)((gy * IMG + gx) * 4);
            unsigned lx = lds_off(xs + i);
            unsigned ly = lds_off(ys + i);
            asm volatile(
                "global_load_async_to_lds_b32 %0, %2, %3\n\t"
                "global_load_async_to_lds_b32 %1, %2, %4"
                :: "v"(lx), "v"(ly), "v"(goff), "s"(xp), "s"(yp)
                : "memory");
        } else {
            xs[i] = 0.f;
            ys[i] = 0.f;
        }
    }
    // zero K-padding rows 22,23 of each of this wave's 5 staging buffers
    #pragma unroll
    for (int q = 0; q < 5; ++q)
        hbuf[((q * 8 + wv) * 24 + 22) * 16 + lane] = 0.f;

    const int m     = lane & 15;          // M (A) / N (B,C) selector
    const int khalf = (lane >> 4) << 1;   // 0 or 2: this lane group's K half
    const int hi8   = (lane >> 4) << 3;   // C-layout row offset
    const int wc0   = wv * 16;            // wave's output column base

    // ---- loop-invariant banded weights: shared by horizontal-B & vertical-A
    // (overlaps with the in-flight async DMA)
    v2f wreg[6];
    #pragma unroll
    for (int j = 0; j < 6; ++j) {
        wreg[j][0] = gw0(4 * j + khalf + 0 - m);
        wreg[j][1] = gw0(4 * j + khalf + 1 - m);
    }

    asm volatile("s_wait_asynccnt 0" ::: "memory");
    __syncthreads();

    // ---- horizontal pass: all 5 quantities share the A-operand loads
    #pragma unroll
    for (int rb = 0; rb < 2; ++rb) {
        const int rowbase = rb * 6;            // rows 0..15 and 6..21
        const int row = rowbase + m;
        const float* xrow = &xs[row * IN_STR + wc0];
        const float* yrow = &ys[row * IN_STR + wc0];
        v8f h0 = {}, h1 = {}, h2 = {}, h3 = {}, h4 = {};
        #pragma unroll
        for (int j = 0; j < 6; ++j) {
            const int col = 4 * j + khalf;     // even -> 8B-aligned v2f load
            v2f xa = *(const v2f*)(xrow + col);
            v2f ya = *(const v2f*)(yrow + col);
            v2f xx = xa * xa, yy = ya * ya, xy = xa * ya;
            h0 = __builtin_amdgcn_wmma_f32_16x16x4_f32(false, xa, false, wreg[j], (short)0, h0, false, false);
            h1 = __builtin_amdgcn_wmma_f32_16x16x4_f32(false, ya, false, wreg[j], (short)0, h1, false, false);
            h2 = __builtin_amdgcn_wmma_f32_16x16x4_f32(false, xx, false, wreg[j], (short)0, h2, false, false);
            h3 = __builtin_amdgcn_wmma_f32_16x16x4_f32(false, yy, false, wreg[j], (short)0, h3, false, false);
            h4 = __builtin_amdgcn_wmma_f32_16x16x4_f32(false, xy, false, wreg[j], (short)0, h4, false, false);
        }
        // C-layout -> LDS staging: row = rowbase + v + hi8, col = m
        #pragma unroll
        for (int v = 0; v < 8; ++v) {
            const int rr16 = (rowbase + v + hi8) * 16 + m;
            hbuf[((0 * 8 + wv) * 24) * 16 + rr16] = h0[v];
            hbuf[((1 * 8 + wv) * 24) * 16 + rr16] = h1[v];
            hbuf[((2 * 8 + wv) * 24) * 16 + rr16] = h2[v];
            hbuf[((3 * 8 + wv) * 24) * 16 + rr16] = h3[v];
            hbuf[((4 * 8 + wv) * 24) * 16 + rr16] = h4[v];
        }
    }
    __syncthreads();

    // ---- vertical pass per quantity: A = banded weights (registers),
    //      B = staged H rows from LDS
    v8f V[5];
    #pragma unroll
    for (int q = 0; q < 5; ++q) {
        const float* hq = &hbuf[(q * 8 + wv) * 24 * 16 + m];
        v8f acc = {};
        #pragma unroll
        for (int j = 0; j < 6; ++j) {
            const int k0 = 4 * j + khalf;
            v2f bm;
            bm[0] = hq[k0 * 16];
            bm[1] = hq[k0 * 16 + 16];
            acc = __builtin_amdgcn_wmma_f32_16x16x4_f32(false, wreg[j], false, bm, (short)0, acc, false, false);
        }
        V[q] = acc;
    }

    // ---- elementwise SSIM map + partial sum (8 outputs per lane)
    const float C1v = 0.0001f, C2v = 0.0009f;
    float s = 0.f;
    #pragma unroll
    for (int e = 0; e < 8; ++e) {
        float mu1 = V[0][e], mu2 = V[1][e];
        float m11 = mu1 * mu1, m22 = mu2 * mu2, m12 = mu1 * mu2;
        float s11 = V[2][e] - m11;
        float s22 = V[3][e] - m22;
        float s12 = V[4][e] - m12;
        float num = (2.f * m12 + C1v) * (2.f * s12 + C2v);
        float den = (m11 + m22 + C1v) * (s11 + s22 + C2v);
        s += num / den;
    }
    red[tid] = s;
    __syncthreads();
    #pragma unroll
    for (int off = 128; off > 0; off >>= 1) {
        if (tid < off) red[tid] += red[tid + off];
        __syncthreads();
    }
    if (tid == 0) partial[blockIdx.x] = red[0];
}

__global__ __launch_bounds__(256) void ssim_reduce_kernel(
    const float* __restrict__ partial, float* __restrict__ out, int n)
{
    __shared__ float red[256];
    float s = 0.f;
    for (int i = threadIdx.x; i < n; i += 256) s += partial[i];
    red[threadIdx.x] = s;
    __syncthreads();
    #pragma unroll
    for (int off = 128; off > 0; off >>= 1) {
        if (threadIdx.x < off) red[threadIdx.x] += red[threadIdx.x + off];
        __syncthreads();
    }
    if (threadIdx.x == 0) out[0] = red[0] * (1.0f / NTOT);
}

extern "C" void kernel_launch(void* const* d_in, const int* in_sizes, int n_in,
                              void* d_out, int out_size, void* d_ws, size_t ws_size,
                              hipStream_t stream) {
    const float* x = (const float*)d_in[0];
    const float* y = (const float*)d_in[1];
    float* out     = (float*)d_out;
    float* partial = (float*)d_ws;     // NBLK floats of scratch

    ssim_tile_kernel<<<dim3(NBLK), dim3(256), 0, stream>>>(x, y, partial);
    ssim_reduce_kernel<<<dim3(1), dim3(256), 0, stream>>>(partial, out, NBLK);
}